// Winograd2x3_53446573032291
// MI455X (gfx1250) — compile-verified
//
#include <hip/hip_runtime.h>

typedef __attribute__((ext_vector_type(2))) float v2f;
typedef __attribute__((ext_vector_type(8))) float v8f;

// ---------------------------------------------------------------------------
// Kernel 1: zero-fill the 63 MB output (bandwidth-bound part of the job).
// ---------------------------------------------------------------------------
__global__ void wg_zero_kernel(float4* __restrict__ out, int n4) {
    int i = blockIdx.x * blockDim.x + threadIdx.x;
    if (i < n4) {
        float4 z; z.x = 0.f; z.y = 0.f; z.z = 0.f; z.w = 0.f;
        out[i] = z;
    }
}

// ---------------------------------------------------------------------------
// Kernel 2: the surviving computation.
//   V[c,u,:] = B^T d B   for d = x[0, c, 0:4, 2u:2u+4],  u in [0,31)
//   U[c,:]   = G g G^T   for g = filters[0, c]
//   M[u,e]   = sum_c U[c,e] * V[c,u,e]          <-- WMMA f32 16x16x4
//   Y        = AT M_u AT^T ; out[b,0,i,2u+l] = Y[i][l]  (broadcast over b)
// One workgroup, 512 threads = 16 wave32s; wave e handles Winograd pos e.
// ---------------------------------------------------------------------------
__global__ __launch_bounds__(512) void wg_compute_kernel(
        const float* __restrict__ x,      // (64,64,64,64)
        const float* __restrict__ flt,    // (64,64,3,3)
        float* __restrict__ out)          // (64,64,62,62)
{
    __shared__ float Vl[64 * 32 * 16];   // [c][u(pad 32)][e]  128 KB
    __shared__ float Ul[64 * 16];        // [c][e]               4 KB
    __shared__ float Ml[32 * 16];        // [u][e]               2 KB

    const int tid = threadIdx.x;

    const float Bm[4][4]  = {{1,0,-1,0},{0,1,1,0},{0,-1,1,0},{0,1,0,-1}};
    const float Gm[4][3]  = {{1.f,0.f,0.f},{0.5f,0.5f,1.2f},{0.5f,-0.5f,0.5f},{0.f,0.f,1.f}};
    const float ATm[2][4] = {{1,1,1,0},{0,1,-1,-1}};

    // ---- Phase 1: input transform V (and filter transform U) into LDS ----
    for (int pair = tid; pair < 64 * 32; pair += 512) {
        const int c = pair >> 5;
        const int u = pair & 31;
        float V[4][4];
        if (u < 31) {
            float d[4][4];
            for (int j = 0; j < 4; ++j)
                for (int k = 0; k < 4; ++k)
                    d[j][k] = x[c * 4096 + j * 64 + 2 * u + k];
            // t = B^T d   (t[i][k] = sum_j Bm[j][i] d[j][k])
            float t[4][4];
            for (int i = 0; i < 4; ++i)
                for (int k = 0; k < 4; ++k) {
                    float s = 0.f;
                    for (int j = 0; j < 4; ++j) s += Bm[j][i] * d[j][k];
                    t[i][k] = s;
                }
            // V = t B
            for (int i = 0; i < 4; ++i)
                for (int l = 0; l < 4; ++l) {
                    float s = 0.f;
                    for (int k = 0; k < 4; ++k) s += t[i][k] * Bm[k][l];
                    V[i][l] = s;
                }
        } else {
            for (int i = 0; i < 4; ++i)
                for (int l = 0; l < 4; ++l) V[i][l] = 0.f;
        }
        for (int i = 0; i < 4; ++i)
            for (int l = 0; l < 4; ++l)
                Vl[(c * 32 + u) * 16 + i * 4 + l] = V[i][l];
    }
    if (tid < 64) {
        const int c = tid;
        float g[3][3];
        for (int j = 0; j < 3; ++j)
            for (int k = 0; k < 3; ++k)
                g[j][k] = flt[c * 9 + j * 3 + k];
        float t[4][3];
        for (int i = 0; i < 4; ++i)
            for (int k = 0; k < 3; ++k) {
                float s = 0.f;
                for (int j = 0; j < 3; ++j) s += Gm[i][j] * g[j][k];
                t[i][k] = s;
            }
        for (int i = 0; i < 4; ++i)
            for (int l = 0; l < 4; ++l) {
                float s = 0.f;
                for (int k = 0; k < 3; ++k) s += t[i][k] * Gm[l][k];
                Ul[c * 16 + i * 4 + l] = s;
            }
    }
    __syncthreads();

    // ---- Phase 2: channel contraction via V_WMMA_F32_16X16X4_F32 ----
    // Wave `e` computes M[u, e] = sum_c V[c,u,e] * U[c,e] for u in its tiles.
    // Fragment layout (ISA 7.12.2, 32-bit 16x4 A / 4x16 B):
    //   (VGPR j, lane L) -> K = 2*(L/16) + j ; M (A) / N (B) = L % 16.
    const int wave = tid >> 5;       // = e, 0..15
    const int lane = tid & 31;
    const int e    = wave;
    const int half = lane >> 4;      // 0: K=0,1   1: K=2,3
    const int row  = lane & 15;      // M row (A) / N col (B)

    for (int ut = 0; ut < 2; ++ut) {
        const int ub = ut * 16;      // u tile base (u=31 padded with zeros)
        v8f acc = {};
        for (int c0 = 0; c0 < 64; c0 += 4) {
            const int cA = c0 + 2 * half;
            v2f a, b;
            a.x = Vl[((cA    ) * 32 + ub + row) * 16 + e];
            a.y = Vl[((cA + 1) * 32 + ub + row) * 16 + e];
            b.x = Ul[(cA    ) * 16 + e];        // same for every N column
            b.y = Ul[(cA + 1) * 16 + e];
            acc = __builtin_amdgcn_wmma_f32_16x16x4_f32(
                    /*neg_a=*/false, a, /*neg_b=*/false, b,
                    /*c_mod=*/(short)0, acc,
                    /*reuse_a=*/false, /*reuse_b=*/false);
        }
        // D[m][n] identical across n; VGPR i @ lane L holds M = i + 8*(L/16).
        if (row == 0) {
            for (int i = 0; i < 8; ++i)
                Ml[(ub + i + 8 * half) * 16 + e] = acc[i];
        }
    }
    __syncthreads();

    // ---- Phase 3: output transform Y = AT M AT^T, broadcast over batch ----
    if (tid < 31) {
        const int u = tid;
        float m4[4][4];
        for (int j = 0; j < 4; ++j)
            for (int k = 0; k < 4; ++k)
                m4[j][k] = Ml[u * 16 + j * 4 + k];
        float t[2][4];
        for (int i = 0; i < 2; ++i)
            for (int k = 0; k < 4; ++k) {
                float s = 0.f;
                for (int j = 0; j < 4; ++j) s += ATm[i][j] * m4[j][k];
                t[i][k] = s;
            }
        float Y[2][2];
        for (int i = 0; i < 2; ++i)
            for (int l = 0; l < 2; ++l) {
                float s = 0.f;
                for (int k = 0; k < 4; ++k) s += t[i][k] * ATm[l][k];
                Y[i][l] = s;
            }
        // out[b, f=0, i, 2u+l];  batch stride 64*62*62 = 246016
        for (int b = 0; b < 64; ++b) {
            const size_t base = (size_t)b * 246016u;
            out[base + 0 * 62 + 2 * u + 0] = Y[0][0];
            out[base + 0 * 62 + 2 * u + 1] = Y[0][1];
            out[base + 1 * 62 + 2 * u + 0] = Y[1][0];
            out[base + 1 * 62 + 2 * u + 1] = Y[1][1];
        }
    }
}

// ---------------------------------------------------------------------------
extern "C" void kernel_launch(void* const* d_in, const int* in_sizes, int n_in,
                              void* d_out, int out_size, void* d_ws, size_t ws_size,
                              hipStream_t stream) {
    (void)in_sizes; (void)n_in; (void)d_ws; (void)ws_size;
    const float* x   = (const float*)d_in[0];   // (64,64,64,64) f32
    const float* flt = (const float*)d_in[1];   // (64,64,3,3)   f32
    float* out = (float*)d_out;                 // (64,64,62,62) f32

    const int n4 = out_size / 4;                // out_size = 15,745,024
    wg_zero_kernel<<<(n4 + 255) / 256, 256, 0, stream>>>((float4*)out, n4);
    wg_compute_kernel<<<1, 512, 0, stream>>>(x, flt, out);
}